// LengthRegularizer_27625229647895
// MI455X (gfx1250) — compile-verified
//
#include <hip/hip_runtime.h>

// LengthRegularizer: B=16, T=512, D=384, MAX_DUR=8, OUT_LEN=3584, pure gather/expand.
// Memory-bound: ~105 MB traffic -> ~4.5us at 23.3 TB/s. No matmul -> no WMMA.
// CDNA5 paths used: async global->LDS staging of csum rows (ASYNCcnt),
// b128 vector ld/st, non-temporal output stores.

namespace {
constexpr int kB       = 16;
constexpr int kT       = 512;
constexpr int kD       = 384;
constexpr int kOutLen  = kT * 7;        // 3584
constexpr int kRowsPerBlock   = 32;     // 8 waves x 4 rows
constexpr int kGatherThreads  = 256;
constexpr int kRowsPerWave    = kRowsPerBlock / (kGatherThreads / 32); // 4
}

typedef float v4f __attribute__((ext_vector_type(4)));

// Exact pointee type the async-LDS builtin wants (from hipcc diagnostic):
//   'int __attribute__((vector_size(2*sizeof(int)))) __device__ *'
typedef int lr_v2i __attribute__((vector_size(2 * sizeof(int))));

#if defined(__HIP_DEVICE_COMPILE__) && defined(__gfx1250__)
#  if __has_builtin(__builtin_amdgcn_global_load_async_to_lds_b64)
#    define LR_ASYNC_B64 1
#  endif
#  if __has_builtin(__builtin_amdgcn_s_wait_asynccnt)
#    define LR_WAIT_ASYNC_BUILTIN 1
#  endif
#endif

__device__ __forceinline__ void lr_wait_async0() {
#if defined(LR_WAIT_ASYNC_BUILTIN)
  __builtin_amdgcn_s_wait_asynccnt(0);
#elif defined(LR_ASYNC_B64)
  asm volatile("s_wait_asynccnt 0" ::: "memory");
#endif
}

// ---------------- Kernel 1: masked inclusive cumsum per batch ----------------
// One block per batch, 512 threads, Hillis-Steele scan in LDS.
__global__ __launch_bounds__(kT)
void lr_scan_kernel(const int* __restrict__ ds, const int* __restrict__ ilens,
                    int* __restrict__ csum) {
  __shared__ int s[kT];
  const int b = blockIdx.x;
  const int t = threadIdx.x;
  const int il = ilens[b];
  s[t] = (t < il) ? ds[b * kT + t] : 0;
  __syncthreads();
  for (int off = 1; off < kT; off <<= 1) {
    const int add = (t >= off) ? s[t - off] : 0;
    __syncthreads();
    s[t] += add;
    __syncthreads();
  }
  csum[b * kT + t] = s[t];
}

// ---------------- Kernel 2: searchsorted gather + pad ----------------
// grid = (OUT_LEN/32, B). Stage csum row (2KB) in LDS via async copy.
// Each wave handles 4 output rows: uniform binary search (broadcast LDS reads),
// then 3x b128 per lane to copy the 1536B feature row; NT stores (write-once).
__global__ __launch_bounds__(kGatherThreads)
void lr_gather_kernel(const float* __restrict__ xs, const int* __restrict__ csum,
                      float* __restrict__ out) {
  __shared__ int s_csum[kT];
  const int b   = blockIdx.y;
  const int tid = threadIdx.x;
  const int* __restrict__ grow = csum + b * kT;

#if defined(LR_ASYNC_B64)
  // 256 lanes x 8B = 2KB: one async b64 per lane, LDS write tracked by ASYNCcnt.
  __builtin_amdgcn_global_load_async_to_lds_b64(
      (__attribute__((address_space(1))) lr_v2i*)(grow + tid * 2),
      (__attribute__((address_space(3))) lr_v2i*)(&s_csum[tid * 2]),
      /*offset=*/0, /*cpol=*/0);
  lr_wait_async0();
#else
  s_csum[tid * 2 + 0] = grow[tid * 2 + 0];
  s_csum[tid * 2 + 1] = grow[tid * 2 + 1];
#endif
  __syncthreads();

  const int total = s_csum[kT - 1];
  const int wave  = tid >> 5;
  const int lane  = tid & 31;
  const int row_base = blockIdx.x * kRowsPerBlock + wave * kRowsPerWave;

#pragma unroll
  for (int r = 0; r < kRowsPerWave; ++r) {
    const int t_out = row_base + r;
    // searchsorted(csum, t_out, side='right'): first i with csum[i] > t_out.
    // Wave-uniform -> LDS broadcast reads, no divergence.
    int lo = 0, hi = kT;
    while (lo < hi) {
      const int mid = (lo + hi) >> 1;
      if (s_csum[mid] <= t_out) lo = mid + 1; else hi = mid;
    }
    const int  idx   = (lo < kT - 1) ? lo : (kT - 1);
    const bool valid = (t_out < total);

    const v4f* __restrict__ src =
        (const v4f*)(xs + ((long)b * kT + idx) * kD);
    v4f* __restrict__ dst =
        (v4f*)(out + ((long)b * kOutLen + t_out) * kD);

#pragma unroll
    for (int i = lane; i < kD / 4; i += 32) {   // 3 iterations of b128
      v4f val = {0.f, 0.f, 0.f, 0.f};
      if (valid) val = src[i];
      __builtin_nontemporal_store(val, dst + i); // write-once stream, TH=NT
    }
  }
}

extern "C" void kernel_launch(void* const* d_in, const int* in_sizes, int n_in,
                              void* d_out, int out_size, void* d_ws, size_t ws_size,
                              hipStream_t stream) {
  (void)in_sizes; (void)n_in; (void)out_size; (void)ws_size;
  const float* xs    = (const float*)d_in[0];
  const int*   ds    = (const int*)  d_in[1];
  const int*   ilens = (const int*)  d_in[2];
  float*       out   = (float*)d_out;
  int*         csum  = (int*)d_ws;   // kB*kT ints = 32 KB of workspace

  lr_scan_kernel<<<kB, kT, 0, stream>>>(ds, ilens, csum);

  dim3 grid(kOutLen / kRowsPerBlock, kB);
  lr_gather_kernel<<<grid, kGatherThreads, 0, stream>>>(xs, csum, out);
}